// GCN_35450660061449
// MI455X (gfx1250) — compile-verified
//
#include <hip/hip_runtime.h>

typedef __attribute__((ext_vector_type(2))) float v2f;
typedef __attribute__((ext_vector_type(8))) float v8f;

#define GCN_N_NODES  50000
#define GCN_N_GRAPHS 64

// ---------------------------------------------------------------- utilities
__global__ void zero_f32(float* __restrict__ p, int n) {
    int i = blockIdx.x * blockDim.x + threadIdx.x;
    if (i < n) p[i] = 0.0f;
}

__global__ void deg_init(float* __restrict__ deg, int n) {
    int i = blockIdx.x * blockDim.x + threadIdx.x;
    if (i < n) deg[i] = 1.0f;                       // self-loop contributes 1
}

__global__ void deg_accum(const int* __restrict__ dst, float* __restrict__ deg, int e) {
    int i = blockIdx.x * blockDim.x + threadIdx.x;
    if (i < e) atomicAdd(&deg[dst[i]], 1.0f);
}

__global__ void deg_finalize(float* __restrict__ deg, int n) {
    int i = blockIdx.x * blockDim.x + threadIdx.x;
    if (i < n) deg[i] = rsqrtf(deg[i]);             // deg >= 1 always
}

// ---------------------------------------------------------------- GEMM (WMMA f32)
// H[M,NCOLS] = A[M,K] @ W[K,NCOLS].
// Block = WAVES waves; all waves share one 16-wide column tile of W staged in
// LDS; each wave owns one 16x16 output tile (16 rows of A).
// Requires M % 16 == 0 (true here: 50000 = 3125*16) -> no load/store guards,
// EXEC stays all-ones around every WMMA.
//
// V_WMMA_F32_16X16X4_F32 layout:
//   A 16x4:  lanes 0-15 -> M=lane, {v0,v1}={K=0,K=1}; lanes 16-31 -> {K=2,K=3}
//   B 4x16:  lanes hold N=lane&15; lanes 0-15 -> {K=0,K=1}; lanes 16-31 -> {K=2,K=3}
//   C/D:     VGPR r -> M = r + (lane<16 ? 0 : 8), N = lane&15
template <int K, int NCOLS, int WAVES>
__global__ void gemm_f32_wmma(const float* __restrict__ A,
                              const float* __restrict__ Wm,
                              float* __restrict__ H, int M) {
    __shared__ float Bs[K * 16];

    const int tid  = threadIdx.x;             // 0 .. 32*WAVES-1
    const int lane = tid & 31;
    const int wav  = tid >> 5;
    const int colBase = blockIdx.y * 16;

    // cooperative stage of the K x 16 B tile (coalesced: consecutive tid -> cols)
    #pragma unroll
    for (int i = tid; i < K * 16; i += 32 * WAVES) {
        const int kk = i >> 4;
        const int cc = i & 15;
        Bs[i] = Wm[kk * NCOLS + colBase + cc];
    }
    __syncthreads();

    const int mtile = blockIdx.x * WAVES + wav;
    if (mtile * 16 < M) {                     // wave-uniform guard
        const int l15 = lane & 15;
        const int hi  = lane >> 4;            // 0: lanes 0-15, 1: lanes 16-31
        const int khi = hi * 2;
        const int mrow = mtile * 16 + l15;

        v8f acc = {0.f, 0.f, 0.f, 0.f, 0.f, 0.f, 0.f, 0.f};
        #pragma unroll 8
        for (int k0 = 0; k0 < K; k0 += 4) {
            v2f a = *(const v2f*)&A[mrow * K + k0 + khi];   // 8B-aligned pair
            v2f b;
            b.x = Bs[(k0 + khi) * 16 + l15];
            b.y = Bs[(k0 + khi + 1) * 16 + l15];
            acc = __builtin_amdgcn_wmma_f32_16x16x4_f32(
                /*neg_a=*/false, a, /*neg_b=*/false, b,
                /*c_mod=*/(short)0, acc, /*reuse_a=*/false, /*reuse_b=*/false);
        }

        const int m0  = mtile * 16 + hi * 8;
        const int col = colBase + l15;
        #pragma unroll
        for (int r = 0; r < 8; ++r)
            H[(m0 + r) * NCOLS + col] = acc[r];
    }
}

// ---------------------------------------------------------------- edge scatter
// One wave32 per edge; F/32 coalesced lanes of gather + float atomic-add.
template <int F>
__global__ void scatter_edges(const float* __restrict__ h,
                              const int* __restrict__ src,
                              const int* __restrict__ dst,
                              const float* __restrict__ dinv,
                              float* __restrict__ agg, int e) {
    const int wid  = (int)((blockIdx.x * blockDim.x + threadIdx.x) >> 5);
    const int lane = threadIdx.x & 31;
    if (wid >= e) return;
    const int s = src[wid];
    __builtin_prefetch(&h[(size_t)s * F + lane], 0, 0);   // global_prefetch
    const int d = dst[wid];
    const float nrm = dinv[s] * dinv[d];
    #pragma unroll
    for (int c = 0; c < F / 32; ++c) {
        float v = h[s * F + c * 32 + lane] * nrm;
        atomicAdd(&agg[d * F + c * 32 + lane], v);
    }
}

// out = act(agg + h*dinv^2 (self-loop) + bias)
template <int F, bool RELU>
__global__ void node_finalize(const float* __restrict__ agg,
                              const float* __restrict__ h,
                              const float* __restrict__ dinv,
                              const float* __restrict__ bias,
                              float* __restrict__ out, int n) {
    const int node = blockIdx.x;
    if (node >= n) return;
    const float di  = dinv[node];
    const float di2 = di * di;
    for (int f = threadIdx.x; f < F; f += blockDim.x) {
        float v = agg[node * F + f] + h[node * F + f] * di2 + bias[f];
        out[node * F + f] = RELU ? fmaxf(v, 0.0f) : v;
    }
}

// ---------------------------------------------------------------- mean pool (F=64)
__global__ void pool_sum(const float* __restrict__ h,
                         const int* __restrict__ batch,
                         float* __restrict__ sums,
                         float* __restrict__ cnts, int n) {
    const int wid  = (int)((blockIdx.x * blockDim.x + threadIdx.x) >> 5);
    const int lane = threadIdx.x & 31;
    if (wid >= n) return;
    const int g = batch[wid];
    if (lane == 0) atomicAdd(&cnts[g], 1.0f);
    atomicAdd(&sums[g * 64 + lane],      h[wid * 64 + lane]);
    atomicAdd(&sums[g * 64 + 32 + lane], h[wid * 64 + 32 + lane]);
}

__global__ void pool_div(const float* __restrict__ sums,
                         const float* __restrict__ cnts,
                         float* __restrict__ out, int n) {
    int i = blockIdx.x * blockDim.x + threadIdx.x;
    if (i < n) out[i] = sums[i] / fmaxf(cnts[i / 64], 1.0f);
}

// ---------------------------------------------------------------- launcher
extern "C" void kernel_launch(void* const* d_in, const int* in_sizes, int n_in,
                              void* d_out, int out_size, void* d_ws, size_t ws_size,
                              hipStream_t stream) {
    const float* x     = (const float*)d_in[0];
    const int*   ei    = (const int*)d_in[1];
    const int*   batch = (const int*)d_in[2];
    const float* Wl[4] = {(const float*)d_in[3], (const float*)d_in[5],
                          (const float*)d_in[7], (const float*)d_in[9]};
    const float* bl[4] = {(const float*)d_in[4], (const float*)d_in[6],
                          (const float*)d_in[8], (const float*)d_in[10]};

    const int N = GCN_N_NODES;              // 50000 (multiple of 16)
    const int E = in_sizes[1] / 2;
    const int* src = ei;        // edge_index[0]
    const int* dst = ei + E;    // edge_index[1]

    auto au = [](size_t v) { return (v + 255) & ~(size_t)255; };
    char* ws = (char*)d_ws;
    size_t off = 0;
    float* deg  = (float*)(ws + off); off += au(sizeof(float) * N);
    float* bufA = (float*)(ws + off); off += au(sizeof(float) * (size_t)N * 96); // h = X@W
    float* bufB = (float*)(ws + off); off += au(sizeof(float) * (size_t)N * 96); // agg
    float* bufC = (float*)(ws + off); off += au(sizeof(float) * (size_t)N * 96); // layer out
    float* sums = (float*)(ws + off); off += au(sizeof(float) * GCN_N_GRAPHS * 64);
    float* cnts = (float*)(ws + off); off += au(sizeof(float) * GCN_N_GRAPHS);
    (void)ws_size; (void)n_in; (void)out_size;

    const int TB = 256;
    // degrees (with self-loop), then in-place rsqrt
    deg_init    <<<(N + TB - 1) / TB, TB, 0, stream>>>(deg, N);
    deg_accum   <<<(E + TB - 1) / TB, TB, 0, stream>>>(dst, deg, E);
    deg_finalize<<<(N + TB - 1) / TB, TB, 0, stream>>>(deg, N);

    constexpr int WV = 4;                   // waves (M-tiles) per GEMM block
    const int MT  = (N + 15) / 16;          // 3125 row tiles
    const int GBX = (MT + WV - 1) / WV;

    // layer 1: 128 -> 96, ReLU
    gemm_f32_wmma<128, 96, WV><<<dim3(GBX, 6), 32 * WV, 0, stream>>>(x, Wl[0], bufA, N);
    zero_f32<<<(N * 96 + TB - 1) / TB, TB, 0, stream>>>(bufB, N * 96);
    scatter_edges<96><<<(E + 7) / 8, 256, 0, stream>>>(bufA, src, dst, deg, bufB, E);
    node_finalize<96, true><<<N, 96, 0, stream>>>(bufB, bufA, deg, bl[0], bufC, N);

    // layers 2,3: 96 -> 96, ReLU
    for (int l = 1; l <= 2; ++l) {
        gemm_f32_wmma<96, 96, WV><<<dim3(GBX, 6), 32 * WV, 0, stream>>>(bufC, Wl[l], bufA, N);
        zero_f32<<<(N * 96 + TB - 1) / TB, TB, 0, stream>>>(bufB, N * 96);
        scatter_edges<96><<<(E + 7) / 8, 256, 0, stream>>>(bufA, src, dst, deg, bufB, E);
        node_finalize<96, true><<<N, 96, 0, stream>>>(bufB, bufA, deg, bl[l], bufC, N);
    }

    // layer 4: 96 -> 64, no activation
    gemm_f32_wmma<96, 64, WV><<<dim3(GBX, 4), 32 * WV, 0, stream>>>(bufC, Wl[3], bufA, N);
    zero_f32<<<(N * 64 + TB - 1) / TB, TB, 0, stream>>>(bufB, N * 64);
    scatter_edges<64><<<(E + 7) / 8, 256, 0, stream>>>(bufA, src, dst, deg, bufB, E);
    node_finalize<64, false><<<N, 64, 0, stream>>>(bufB, bufA, deg, bl[3], bufC, N);

    // global mean pool
    zero_f32<<<(GCN_N_GRAPHS * 64 + TB - 1) / TB, TB, 0, stream>>>(sums, GCN_N_GRAPHS * 64);
    zero_f32<<<1, 64, 0, stream>>>(cnts, GCN_N_GRAPHS);
    pool_sum<<<(N + 7) / 8, 256, 0, stream>>>(bufC, batch, sums, cnts, N);
    pool_div<<<(GCN_N_GRAPHS * 64 + TB - 1) / TB, TB, 0, stream>>>(sums, cnts, (float*)d_out,
                                                                   GCN_N_GRAPHS * 64);
}